// S4ConvKernel_38027640438873
// MI455X (gfx1250) — compile-verified
//
#include <hip/hip_runtime.h>

typedef float v2f __attribute__((ext_vector_type(2)));
typedef float v8f __attribute__((ext_vector_type(8)));

#define N_ST   32          // states per row (n)
#define KDIM   64          // 2*n (Re/Im interleaved contraction dim)
#define L_DIM  4096
#define NTILES 16          // L / 256
#define WAVES  4

__global__ __launch_bounds__(32 * WAVES)
void s4_vandermonde_wmma(const float* __restrict__ log_dt,
                         const float* __restrict__ C_re,
                         const float* __restrict__ C_im,
                         const float* __restrict__ log_A_re,
                         const float* __restrict__ A_im,
                         float* __restrict__ out)
{
    __shared__ float As[16 * KDIM];           // A = W16^j, shared by all waves of this h
    __shared__ float Bs[WAVES][16 * KDIM];    // per-wave B staging

    const int h    = blockIdx.x;
    const int lane = threadIdx.x & 31;        // = state index n
    const int wv   = threadIdx.x >> 5;
    const int lo   = lane & 15;
    const int hi   = lane >> 4;

    // ---------------- per-state setup (transcendentals only here) --------------
    const float dt  = __expf(log_dt[h]);
    const float cre = C_re[h * N_ST + lane];
    const float cim = C_im[h * N_ST + lane];
    const float Are = -__expf(log_A_re[h * N_ST + lane]);
    const float Aim = A_im[h * N_ST + lane];
    const float ar  = dt * Are;
    const float ai  = dt * Aim;
    const float er  = __expf(ar);
    const float wr  = er * __cosf(ai);
    const float wi  = er * __sinf(ai);

    // g = 2 * C * (w - 1) / A   (complex)
    const float numr = wr - 1.0f, numi = wi;
    const float tr   = cre * numr - cim * numi;
    const float ti   = cre * numi + cim * numr;
    const float inv  = 1.0f / (Are * Are + Aim * Aim);
    float gr = 2.0f * (tr * Are + ti * Aim) * inv;
    float gi = 2.0f * (ti * Are - tr * Aim) * inv;

    // W16 = w^16, W256 = w^256, W1024 = w^1024 (repeated squaring)
    float w16r = wr, w16i = wi;
#pragma unroll
    for (int s = 0; s < 4; ++s) { float r = w16r * w16r - w16i * w16i; w16i = 2.0f * w16r * w16i; w16r = r; }
    float w256r = w16r, w256i = w16i;
#pragma unroll
    for (int s = 0; s < 4; ++s) { float r = w256r * w256r - w256i * w256i; w256i = 2.0f * w256r * w256i; w256r = r; }
    float w1024r = w256r, w1024i = w256i;
#pragma unroll
    for (int s = 0; s < 2; ++s) { float r = w1024r * w1024r - w1024i * w1024i; w1024i = 2.0f * w1024r * w1024i; w1024r = r; }

    // ---------------- A matrix: As[j][2n]=Re(W16^j), As[j][2n+1]=Im(W16^j) -----
    if (wv == 0) {
        float pr = 1.0f, pi = 0.0f;
#pragma unroll
        for (int j = 0; j < 16; ++j) {
            As[j * KDIM + 2 * lane]     = pr;
            As[j * KDIM + 2 * lane + 1] = pi;
            float r = pr * w16r - pi * w16i;
            pi = pr * w16i + pi * w16r;
            pr = r;
        }
    }
    __syncthreads();

    // WMMA fragment addressing (A and B use the same pattern):
    //   value v of fragment kb on lane L = buf[(L%16)*KDIM + kb*4 + 2*(L/16) + v]
    const int fragOff = lo * KDIM + 2 * hi;

    v2f afrag[16];
#pragma unroll
    for (int kb = 0; kb < 16; ++kb) {
        afrag[kb].x = As[fragOff + kb * 4];
        afrag[kb].y = As[fragOff + kb * 4 + 1];
    }

    // g_t start for this wave: g * W256^wv  (wave wv owns tiles t = wv + 4k)
    float grt = gr, git = gi;
    for (int u = 0; u < wv; ++u) {
        float r = grt * w256r - git * w256i;
        git = grt * w256i + git * w256r;
        grt = r;
    }

    float* __restrict__ outh = out + (size_t)h * L_DIM;

    for (int k = 0; k < NTILES / WAVES; ++k) {
        const int t = wv + WAVES * k;

        // B matrix: Bs[l0][2n]=Re(q), Bs[l0][2n+1]=-Im(q), q = g_t * w^{l0}
        float qr = grt, qi = git;
#pragma unroll
        for (int l = 0; l < 16; ++l) {
            Bs[wv][l * KDIM + 2 * lane]     = qr;
            Bs[wv][l * KDIM + 2 * lane + 1] = -qi;
            float r = qr * wr - qi * wi;
            qi = qr * wi + qi * wr;
            qr = r;
        }
        __syncthreads();

        v8f acc = {};
#pragma unroll
        for (int kb = 0; kb < 16; ++kb) {
            v2f bfrag;
            bfrag.x = Bs[wv][fragOff + kb * 4];
            bfrag.y = Bs[wv][fragOff + kb * 4 + 1];
            acc = __builtin_amdgcn_wmma_f32_16x16x4_f32(
                /*neg_a=*/false, afrag[kb],
                /*neg_b=*/false, bfrag,
                /*c_mod=*/(short)0, acc,
                /*reuse_a=*/false, /*reuse_b=*/false);
        }
        __syncthreads();

        // D layout: lane L, vgpr j -> M = j + 8*(L/16), N = L%16
        // out element l = t*256 + M*16 + N
#pragma unroll
        for (int j = 0; j < 8; ++j) {
            outh[t * 256 + (j + 8 * hi) * 16 + lo] = acc[j];
        }

        // advance g_t by W1024 (4 tiles ahead)
        float r = grt * w1024r - git * w1024i;
        git = grt * w1024i + git * w1024r;
        grt = r;
    }
}

extern "C" void kernel_launch(void* const* d_in, const int* in_sizes, int n_in,
                              void* d_out, int out_size, void* d_ws, size_t ws_size,
                              hipStream_t stream) {
    const float* log_dt   = (const float*)d_in[0];
    const float* C_re     = (const float*)d_in[1];
    const float* C_im     = (const float*)d_in[2];
    const float* log_A_re = (const float*)d_in[3];
    const float* A_im     = (const float*)d_in[4];
    float* out = (float*)d_out;

    const int H = in_sizes[0];   // 1024; L fixed at 4096, n fixed at 32 per reference
    s4_vandermonde_wmma<<<H, 32 * WAVES, 0, stream>>>(log_dt, C_re, C_im, log_A_re, A_im, out);
}